// CopyGenerator_63780264345832
// MI455X (gfx1250) — compile-verified
//
#include <hip/hip_runtime.h>
#include <math.h>

typedef float v2f __attribute__((ext_vector_type(2)));
typedef float v8f __attribute__((ext_vector_type(8)));

#define COPY_IDX 4
#define PAD_IDX  0
#define EPS_VAL  1e-10f

#define WMMA_F32(A, B, C) \
    __builtin_amdgcn_wmma_f32_16x16x4_f32(false, (A), false, (B), (short)0, (C), false, false)

// ---------------------------------------------------------------------------
// Kernel 1: logits = hidden @ W^T + bias  via V_WMMA_F32_16X16X4_F32.
// hidden: [B,H] row-major (A matrix, 16x4 frags)
// W:      [V,H] row-major -> B matrix (4x16) column n == W row n, so both A
//         and B fragments are contiguous float2 loads along H.
//
// Each wave computes a 32(M) x 32(N) tile: 4 accumulators from the outer
// product of 2 A-fragments x 2 B-fragments (4 WMMA per 4 loads per K-step of
// 4). Fragments are double-buffered: next K-step's loads are issued before
// the WMMAs consuming the current K-step, so the matrix pipe hides load
// latency instead of draining loadcnt each step.
//
// Block = 8 waves arranged 2(M) x 4(N) -> 64 x 128 block tile.
// Assumes B % 64 == 0, H % 8 == 0; V only needs V % 16 == 0 (guarded).
// ---------------------------------------------------------------------------
__global__ __launch_bounds__(256) void k_gemm_logits(
    const float* __restrict__ hidden,
    const float* __restrict__ W,
    const float* __restrict__ bias,
    float* __restrict__ logits,
    int B, int H, int V)
{
    const int lane = threadIdx.x & 31;
    const int wave = threadIdx.x >> 5;
    const int half = lane >> 4;   // 0: lanes 0-15 (K=0,1) ; 1: lanes 16-31 (K=2,3)
    const int r    = lane & 15;

    const int n0 = blockIdx.x * 128 + (wave & 3) * 32;  // first 16-col group
    const int m0 = blockIdx.y * 64 + (wave >> 2) * 32;  // first 16-row group
    if (n0 >= V) return;
    const int  n1   = n0 + 16;                          // second 16-col group
    const bool n1ok = (n1 < V);

    v8f acc00 = {};  // (m0..m0+15)  x (n0..n0+15)
    v8f acc01 = {};  // (m0..m0+15)  x (n1..n1+15)
    v8f acc10 = {};  // (m0+16..+31) x (n0..n0+15)
    v8f acc11 = {};  // (m0+16..+31) x (n1..n1+15)

    const float* aptr0 = hidden + (size_t)(m0 + r)      * H + 2 * half;
    const float* aptr1 = hidden + (size_t)(m0 + 16 + r) * H + 2 * half;
    const float* bptr0 = W + (size_t)(n0 + r) * H + 2 * half;
    const float* bptr1 = W + (size_t)((n1ok ? n1 : n0) + r) * H + 2 * half;

    // Prologue: K-step 0 fragments.
    v2f a0 = *(const v2f*)(aptr0);
    v2f a1 = *(const v2f*)(aptr1);
    v2f b0 = *(const v2f*)(bptr0);
    v2f b1 = *(const v2f*)(bptr1);

    #pragma unroll 2
    for (int k = 4; k < H; k += 4) {
        // Issue next K-step's loads first ...
        v2f na0 = *(const v2f*)(aptr0 + k);
        v2f na1 = *(const v2f*)(aptr1 + k);
        v2f nb0 = *(const v2f*)(bptr0 + k);
        v2f nb1 = *(const v2f*)(bptr1 + k);
        // ... then consume the previous step's registers.
        acc00 = WMMA_F32(a0, b0, acc00);
        acc01 = WMMA_F32(a0, b1, acc01);
        acc10 = WMMA_F32(a1, b0, acc10);
        acc11 = WMMA_F32(a1, b1, acc11);
        a0 = na0; a1 = na1; b0 = nb0; b1 = nb1;
    }
    // Epilogue: last K-step.
    acc00 = WMMA_F32(a0, b0, acc00);
    acc01 = WMMA_F32(a0, b1, acc01);
    acc10 = WMMA_F32(a1, b0, acc10);
    acc11 = WMMA_F32(a1, b1, acc11);

    // C/D layout: VGPR j -> M = j + 8*half, N = lane&15.
    const float bv0 = bias[n0 + r];
    const float bv1 = n1ok ? bias[n1 + r] : 0.f;
    #pragma unroll
    for (int j = 0; j < 8; ++j) {
        const int m = m0 + j + 8 * half;
        logits[(size_t)m        * V + n0 + r] = acc00[j] + bv0;
        logits[(size_t)(m + 16) * V + n0 + r] = acc10[j] + bv0;
        if (n1ok) {
            logits[(size_t)m        * V + n1 + r] = acc01[j] + bv1;
            logits[(size_t)(m + 16) * V + n1 + r] = acc11[j] + bv1;
        }
    }
}

// ---------------------------------------------------------------------------
// Kernel 2: per-row online softmax stats over mod_logits (COPY pinned to
// 1e-10) + sigmoid copy gate. One 256-thread block per row.
// ---------------------------------------------------------------------------
__global__ __launch_bounds__(256) void k_rowstats(
    const float* __restrict__ logits,
    float* __restrict__ rowmax,
    float* __restrict__ rowsum,
    float* __restrict__ copygate,
    int V)
{
    const int b = blockIdx.x;
    const int tid = threadIdx.x;
    const float* row = logits + (size_t)b * V;

    float m = -INFINITY, s = 0.f;
    for (int v = tid; v < V; v += blockDim.x) {
        float x = (v == COPY_IDX) ? 1e-10f : row[v];
        float nm = fmaxf(m, x);
        s = s * __expf(m - nm) + __expf(x - nm);
        m = nm;
    }

    __shared__ float sm[256];
    __shared__ float ss[256];
    sm[tid] = m; ss[tid] = s;
    __syncthreads();
    for (int o = 128; o > 0; o >>= 1) {
        if (tid < o) {
            float m2 = sm[tid + o], s2 = ss[tid + o];
            float M  = fmaxf(sm[tid], m2);
            ss[tid]  = ss[tid] * __expf(sm[tid] - M) + s2 * __expf(m2 - M);
            sm[tid]  = M;
        }
        __syncthreads();
    }
    if (tid == 0) {
        rowmax[b] = sm[0];
        rowsum[b] = ss[0];
        float lc  = row[COPY_IDX];
        copygate[b] = 1.f / (1.f + __expf(-lc));
    }
}

// ---------------------------------------------------------------------------
// Kernel 3: out[b,v] = softmax(mod_logits)[b,v] * (1 - copy[b])
// ---------------------------------------------------------------------------
__global__ __launch_bounds__(256) void k_prob(
    const float* __restrict__ logits,
    const float* __restrict__ rowmax,
    const float* __restrict__ rowsum,
    const float* __restrict__ copygate,
    float* __restrict__ out,
    int V, long long total)
{
    long long i = (long long)blockIdx.x * blockDim.x + threadIdx.x;
    if (i >= total) return;
    int b = (int)(i / V);
    int v = (int)(i - (long long)b * V);
    float x = (v == COPY_IDX) ? 1e-10f : logits[i];
    float p = __expf(x - rowmax[b]) / rowsum[b];
    out[i] = p * (1.f - copygate[b]);
}

// ---------------------------------------------------------------------------
// Kernel 4: batched scatter-add of attn*copy through alignment[src].
// ---------------------------------------------------------------------------
__global__ __launch_bounds__(256) void k_scatter(
    const int* __restrict__ src,
    const float* __restrict__ attn,
    const int* __restrict__ alignment,
    const float* __restrict__ copygate,
    float* __restrict__ out,
    int S, int V, int total)
{
    int i = blockIdx.x * blockDim.x + threadIdx.x;
    if (i >= total) return;
    int b = i / S;
    int tgt = alignment[src[i]];
    atomicAdd(out + (size_t)b * V + tgt, attn[i] * copygate[b]);
}

// ---------------------------------------------------------------------------
// Kernel 5: PAD <- EPS, renormalize row, log. One block per row.
// ---------------------------------------------------------------------------
__global__ __launch_bounds__(256) void k_final(
    float* __restrict__ out, int V)
{
    const int b = blockIdx.x;
    const int tid = threadIdx.x;
    float* row = out + (size_t)b * V;

    if (tid == 0) row[PAD_IDX] = EPS_VAL;
    __syncthreads();

    float s = 0.f;
    for (int v = tid; v < V; v += blockDim.x) s += row[v];

    __shared__ float ss[256];
    ss[tid] = s;
    __syncthreads();
    for (int o = 128; o > 0; o >>= 1) {
        if (tid < o) ss[tid] += ss[tid + o];
        __syncthreads();
    }
    const float inv_norm = 1.f / ss[0];

    for (int v = tid; v < V; v += blockDim.x)
        row[v] = __logf(row[v] * inv_norm + EPS_VAL);
}

// ---------------------------------------------------------------------------
// Host launcher.
// Inputs (setup_inputs order): hidden[B,H] f32, src[B,S] int, attn[B,S] f32,
// W[V,H] f32, b[V] f32, alignment[SRC_V] int.  d_out: [B,V] f32.
// d_ws: staging for logits (B*V f32) + row stats (3*B f32) ~ 205 MB.
// ---------------------------------------------------------------------------
extern "C" void kernel_launch(void* const* d_in, const int* in_sizes, int n_in,
                              void* d_out, int out_size, void* d_ws, size_t ws_size,
                              hipStream_t stream) {
    const float* hidden    = (const float*)d_in[0];
    const int*   src       = (const int*)  d_in[1];
    const float* attn      = (const float*)d_in[2];
    const float* W         = (const float*)d_in[3];
    const float* bias      = (const float*)d_in[4];
    const int*   alignment = (const int*)  d_in[5];

    const int V = in_sizes[4];
    const int H = in_sizes[3] / V;
    const int B = in_sizes[0] / H;
    const int S = in_sizes[1] / B;

    float* logits   = (float*)d_ws;
    float* rowmax   = logits + (size_t)B * V;
    float* rowsum   = rowmax + B;
    float* copygate = rowsum + B;
    float* out      = (float*)d_out;

    // 1. GEMM (WMMA f32): 64x128 block tile, 32x32 wave tile.
    dim3 ggrid((V + 127) / 128, (B + 63) / 64);
    k_gemm_logits<<<ggrid, 256, 0, stream>>>(hidden, W, bias, logits, B, H, V);

    // 2. Row stats
    k_rowstats<<<B, 256, 0, stream>>>(logits, rowmax, rowsum, copygate, V);

    // 3. Prob init
    const long long total = (long long)B * V;
    const int nb = (int)((total + 255) / 256);
    k_prob<<<nb, 256, 0, stream>>>(logits, rowmax, rowsum, copygate, out, V, total);

    // 4. Scatter-add
    const int ts = B * S;
    k_scatter<<<(ts + 255) / 256, 256, 0, stream>>>(src, attn, alignment, copygate, out, S, V, ts);

    // 5. Finalize
    k_final<<<B, 256, 0, stream>>>(out, V);
}